// Q_MultiHeadedAttention_67559835566275
// MI455X (gfx1250) — compile-verified
//
#include <hip/hip_runtime.h>

// Q_MultiHeadedAttention for MI455X (gfx1250), exact-math reduction:
//   phi_i   = (pi/64) * sum_p (-1)^pc(p) |E_q x_q,i|_p^2 / |x_q,i|^2      (key drops out!)
//   u_j     = E_v x_v,j / |x_v,j|                                         (rank-1 rho)
//   w_ij[m] = |(B(phi_i) u_j)[perm(m)]|^2 ;  B = RyL(phi/2) P_ring RxL(phi/2)
//   head[i] = ZS1 * invWHT8( prod_j WHT8(w_ij) )                          (XOR-conv fold)
// Heavy step: per (h,i) a real-represented complex matmul (16x16)x(16x128)
// -> V_WMMA_F32_16X16X4_F32 (full fp32 precision), 32 WMMAs per wave.
// All operands are preloaded into registers (b64 clause) so the WMMA chain
// runs without per-instruction load waits.

#define D  8
#define S  128
#define H  2

typedef __attribute__((ext_vector_type(2))) float v2f;
typedef __attribute__((ext_vector_type(4))) float v4f;
typedef __attribute__((ext_vector_type(8))) float v8f;

__device__ __forceinline__ int ring_f(int j) {
  // P_RING forward map (qubit0 = MSB): CNOT(0->1), CNOT(1->2), CNOT(2->0)
  int b2 = (j >> 2) & 1, b1 = (j >> 1) & 1, b0 = j & 1;
  int nb1 = b1 ^ b2;
  int nb0 = b0 ^ nb1;
  int nb2 = b2 ^ nb0;
  return (nb2 << 2) | (nb1 << 1) | nb0;
}
__device__ __forceinline__ float zsign(int p) { return (__popc(p) & 1) ? -1.f : 1.f; }

__device__ __forceinline__ void wht8(float w[8]) {
#define QBFLY(i0, i1) { float u_ = w[i0], v_ = w[i1]; w[i0] = u_ + v_; w[i1] = u_ - v_; }
  QBFLY(0,1) QBFLY(2,3) QBFLY(4,5) QBFLY(6,7)
  QBFLY(0,2) QBFLY(1,3) QBFLY(4,6) QBFLY(5,7)
  QBFLY(0,4) QBFLY(1,5) QBFLY(2,6) QBFLY(3,7)
#undef QBFLY
}

// ---------------- Kernel A: per-head setup -------------------------------
// grid = H blocks x 128 threads. Builds:
//   Xt[h][S][16] : per column j, [Re(u_j)[0..7], Im(u_j)[0..7]]  (b64-load friendly)
//   Amat[h][i][16][16] : [[Br,-Bi],[Bi,Br]] for B(phi_i)
__global__ void qmha_setup(const float* __restrict__ xq_all,
                           const float* __restrict__ xv_all,
                           const float* __restrict__ wq,
                           const float* __restrict__ wv,
                           float* __restrict__ Xt,
                           float* __restrict__ Amat) {
  const int h = blockIdx.x;
  const int tid = threadIdx.x;
  __shared__ float Er[2][64], Ei[2][64];      // 0 = E_q, 1 = E_v (8x8 complex)
  __shared__ float Rr[2][3][4], Ri[2][3][4];  // per-qubit 2x2 rotations

  const float W_MUL = 0.63245553203367586640f;  // sqrt(2)*5^-0.5
  if (tid == 0) {
    const float* wsel[2] = { wq + h * 9, wv + h * 9 };
    for (int e = 0; e < 2; e++)
      for (int qb = 0; qb < 3; qb++) {
        float tx = wsel[e][qb * 3 + 0] * W_MUL;
        float ty = wsel[e][qb * 3 + 1] * W_MUL;
        float tz = wsel[e][qb * 3 + 2] * W_MUL;
        float cx = __cosf(tx * 0.5f), sx = __sinf(tx * 0.5f);
        float cy = __cosf(ty * 0.5f), sy = __sinf(ty * 0.5f);
        // M = Ry * Rx ; Rx=[[c,-is],[-is,c]], Ry=[[c,-s],[s,c]]
        float mr[4], mi[4];
        mr[0] =  cy * cx; mi[0] =  sy * sx;
        mr[1] = -sy * cx; mi[1] = -cy * sx;
        mr[2] =  sy * cx; mi[2] = -cy * sx;
        mr[3] =  cy * cx; mi[3] = -sy * sx;
        // R = Rz * M ; Rz = diag(e^{-i tz/2}, e^{+i tz/2})
        float cz = __cosf(tz * 0.5f), sz = __sinf(tz * 0.5f);
        Rr[e][qb][0] = mr[0] * cz + mi[0] * sz; Ri[e][qb][0] = mi[0] * cz - mr[0] * sz;
        Rr[e][qb][1] = mr[1] * cz + mi[1] * sz; Ri[e][qb][1] = mi[1] * cz - mr[1] * sz;
        Rr[e][qb][2] = mr[2] * cz - mi[2] * sz; Ri[e][qb][2] = mi[2] * cz + mr[2] * sz;
        Rr[e][qb][3] = mr[3] * cz - mi[3] * sz; Ri[e][qb][3] = mi[3] * cz + mr[3] * sz;
      }
  }
  __syncthreads();
  if (tid < 64) {
    // E = P_RING @ (R0 kron R1 kron R2); qubit0 -> index MSB
    int j = tid >> 3, c = tid & 7;
    int j2 = (j >> 2) & 1, j1 = (j >> 1) & 1, j0 = j & 1;
    int c2 = (c >> 2) & 1, c1 = (c >> 1) & 1, c0 = c & 1;
    int dst = ring_f(j) * 8 + c;
    for (int e = 0; e < 2; e++) {
      float ar = Rr[e][0][j2 * 2 + c2], ai = Ri[e][0][j2 * 2 + c2];
      float br = Rr[e][1][j1 * 2 + c1], bi = Ri[e][1][j1 * 2 + c1];
      float cr = Rr[e][2][j0 * 2 + c0], ci = Ri[e][2][j0 * 2 + c0];
      float tr = ar * br - ai * bi, ti = ar * bi + ai * br;
      Er[e][dst] = tr * cr - ti * ci;
      Ei[e][dst] = tr * ci + ti * cr;
    }
  }
  __syncthreads();

  const int i = tid;  // sequence position 0..127
  // --- phi_i from query row (key input provably cancels: Tr(rho_k)=1) ---
  const float* xq = xq_all + i * D;
  float nq = 0.f;
#pragma unroll
  for (int c = 0; c < 8; c++) nq += xq[c] * xq[c];
  float qsum = 0.f;
#pragma unroll
  for (int p = 0; p < 8; p++) {
    float vr = 0.f, vi = 0.f;
#pragma unroll
    for (int c = 0; c < 8; c++) { vr += Er[0][p * 8 + c] * xq[c]; vi += Ei[0][p * 8 + c] * xq[c]; }
    qsum += zsign(p) * (vr * vr + vi * vi);
  }
  const float PI_F = 3.14159265358979323846f;
  float phi = PI_F * qsum / (64.f * nq);

  // --- u_j = E_v x_v / |x_v|  -> Xt[h][j][0..15] = [Re u, Im u] ---
  const float* xv = xv_all + i * D;
  float nv = 0.f;
#pragma unroll
  for (int c = 0; c < 8; c++) nv += xv[c] * xv[c];
  float invs = rsqrtf(nv);
  float* xo = Xt + (size_t)(h * S + i) * 16;
#pragma unroll
  for (int p = 0; p < 8; p++) {
    float ur = 0.f, ui = 0.f;
#pragma unroll
    for (int c = 0; c < 8; c++) { ur += Er[1][p * 8 + c] * xv[c]; ui += Ei[1][p * 8 + c] * xv[c]; }
    xo[p]     = ur * invs;
    xo[8 + p] = ui * invs;
  }

  // --- B(phi) = RyL * P_RING * RxL (Rz layer is diagonal -> drops out) ---
  float c4 = __cosf(phi * 0.25f), s4 = __sinf(phi * 0.25f);
  float cp[4] = {1.f, c4, c4 * c4, c4 * c4 * c4};
  float sp[4] = {1.f, s4, s4 * s4, s4 * s4 * s4};
  const float rxr[4] = {1.f, 0.f, -1.f, 0.f};   // (-i)^w real part
  const float rxi[4] = {0.f, -1.f, 0.f, 1.f};   // (-i)^w imag part
  float M1r[64], M1i[64];  // P_RING * RxL
#pragma unroll
  for (int a = 0; a < 8; a++) {
    int fa = ring_f(a);
#pragma unroll
    for (int b = 0; b < 8; b++) {
      int w = __popc(a ^ b);
      float mag = cp[3 - w] * sp[w];
      M1r[fa * 8 + b] = mag * rxr[w];
      M1i[fa * 8 + b] = mag * rxi[w];
    }
  }
  float* Ab = Amat + (size_t)(h * S + i) * 256;
#pragma unroll
  for (int m = 0; m < 8; m++) {
    float ry[8];
#pragma unroll
    for (int a = 0; a < 8; a++) {
      int w = __popc(m ^ a);
      float val = cp[3 - w] * sp[w];
      ry[a] = (__popc(a & ~m) & 1) ? -val : val;   // Ry entry signs
    }
#pragma unroll
    for (int b = 0; b < 8; b++) {
      float br = 0.f, bi = 0.f;
#pragma unroll
      for (int a = 0; a < 8; a++) { br += ry[a] * M1r[a * 8 + b]; bi += ry[a] * M1i[a * 8 + b]; }
      Ab[m * 16 + b]            = br;   // [[Br, -Bi],
      Ab[m * 16 + 8 + b]        = -bi;  //  [Bi,  Br]]
      Ab[(8 + m) * 16 + b]      = bi;
      Ab[(8 + m) * 16 + 8 + b]  = br;
    }
  }
}

// ---------------- Kernel B: WMMA core + WHT fold -------------------------
// grid = H*S blocks x 32 threads (one wave, EXEC all ones for WMMA).
// Per block (h,i): C(16xS) = A_i(16x16) @ X(16xS) via v_wmma_f32_16x16x4_f32,
// then w[m,j] = Cr^2 + Ci^2 (cross-half shfl), row-perm, WHT8, product over j,
// inverse WHT8, Z-sign -> hx[i][h*8+m].
__global__ void qmha_main(const float* __restrict__ Xt,
                          const float* __restrict__ Amat,
                          float* __restrict__ hx) {
  const int h = blockIdx.x >> 7;
  const int i = blockIdx.x & 127;
  const int lane = threadIdx.x;
  const int n  = lane & 15;   // N index within tile; also M index for A operand
  const int kh = lane >> 4;   // K half-select per 32-bit operand layout

  // ---- Preload all WMMA operands as aligned b64 fragments (one clause) ----
  // A operand: lanes 0-15 M=0..15 (lanes 16-31 same M, K+2); VGPR0 K=0/2, VGPR1 K=1/3
  const float* Ab = Amat + (size_t)(h * S + i) * 256;
  v2f a[4];
#pragma unroll
  for (int kk = 0; kk < 4; kk++) {
    int k0 = 4 * kk + 2 * kh;
    a[kk] = *(const v2f*)(Ab + n * 16 + k0);
  }
  // B operand fragments for all 8 N-tiles: X column-major rows K=k0,k0+1
  const float* Xb = Xt + (size_t)h * S * 16;
  v2f bq[8][4];
#pragma unroll
  for (int t = 0; t < 8; t++) {
    int col = t * 16 + n;
#pragma unroll
    for (int kk = 0; kk < 4; kk++) {
      int k0 = 4 * kk + 2 * kh;
      bq[t][kk] = *(const v2f*)(Xb + col * 16 + k0);
    }
  }

  float prod[8] = {1.f, 1.f, 1.f, 1.f, 1.f, 1.f, 1.f, 1.f};

#pragma unroll
  for (int t = 0; t < 8; t++) {
    v8f c = {};
#pragma unroll
    for (int kk = 0; kk < 4; kk++)
      c = __builtin_amdgcn_wmma_f32_16x16x4_f32(false, a[kk], false, bq[t][kk],
                                                (short)0, c, false, false);
    // D layout: VGPR vr -> (M=vr, lanes0-15) / (M=vr+8, lanes16-31).
    // Rows 0-7 = Re(Bu), rows 8-15 = Im(Bu): combine across wave halves.
    float w[8];
#pragma unroll
    for (int vr = 0; vr < 8; vr++) {
      float x0 = c[vr];
      float x1 = __shfl_xor(x0, 16, 32);
      w[vr] = x0 * x0 + x1 * x1;     // |(B u_j)[vr]|^2, duplicated per half
    }
    // measurement-basis permutation from leading P_RING of the _uv circuit
    float wp[8];
#pragma unroll
    for (int j = 0; j < 8; j++) wp[ring_f(j)] = w[j];
    wht8(wp);                        // XOR-conv diagonalization
#pragma unroll
    for (int m = 0; m < 8; m++) prod[m] *= wp[m];
  }

  // product across the 16 distinct columns handled by this half-wave
#pragma unroll
  for (int off = 1; off < 16; off <<= 1) {
#pragma unroll
    for (int m = 0; m < 8; m++) prod[m] *= __shfl_xor(prod[m], off, 32);
  }

  wht8(prod);                        // inverse WHT (x 1/8)
  if (lane == 0) {
    v4f lo, hi;
#pragma unroll
    for (int m = 0; m < 4; m++) lo[m] = zsign(m) * prod[m] * 0.125f;
#pragma unroll
    for (int m = 4; m < 8; m++) hi[m - 4] = zsign(m) * prod[m] * 0.125f;
    float* dst = hx + i * 16 + h * 8;      // 32B aligned
    *(v4f*)dst       = lo;
    *(v4f*)(dst + 4) = hi;
  }
}

// ---------------- Kernel C: output linear layer --------------------------
__global__ void qmha_out(const float* __restrict__ hx,
                         const float* __restrict__ lw,
                         const float* __restrict__ lb,
                         float* __restrict__ y) {
  int t = blockIdx.x * blockDim.x + threadIdx.x;
  if (t >= S * D) return;
  int s = t >> 3, d = t & 7;
  float acc = lb[d];
#pragma unroll
  for (int c = 0; c < 16; c++) acc += hx[s * 16 + c] * lw[d * 16 + c];
  y[t] = acc;
}

extern "C" void kernel_launch(void* const* d_in, const int* in_sizes, int n_in,
                              void* d_out, int out_size, void* d_ws, size_t ws_size,
                              hipStream_t stream) {
  const float* q  = (const float*)d_in[0];
  // d_in[1] (key) and d_in[4] (wk) cancel analytically: sum_q diag(rho_k)=Tr=1.
  const float* v  = (const float*)d_in[2];
  const float* wq = (const float*)d_in[3];
  const float* wv = (const float*)d_in[5];
  const float* lw = (const float*)d_in[6];
  const float* lb = (const float*)d_in[7];

  float* ws   = (float*)d_ws;
  float* Xt   = ws;                  // H*S*16          = 4096 floats
  float* Amat = ws + 4096;           // H*S*16*16       = 65536 floats
  float* hx   = ws + 4096 + 65536;   // S*16            = 2048 floats

  qmha_setup<<<H, S, 0, stream>>>(q, v, wq, wv, Xt, Amat);
  qmha_main<<<H * S, 32, 0, stream>>>(Xt, Amat, hx);
  qmha_out<<<(S * D + 255) / 256, 256, 0, stream>>>(hx, lw, lb, (float*)d_out);
}